// OptimizedGraphAutoencoder_88433376624803
// MI455X (gfx1250) — compile-verified
//
#include <hip/hip_runtime.h>
#include <hip/hip_bf16.h>

typedef __attribute__((ext_vector_type(2))) float v2f;
typedef __attribute__((ext_vector_type(8))) float v8f;
typedef __attribute__((ext_vector_type(4))) int v4i;

static constexpr int NN = 16384;   // nodes
static constexpr int EE = 262144;  // edges
static constexpr int GG = 128;     // graphs
static constexpr int NF = 38;      // node features

#if defined(__gfx1250__) && __has_builtin(__builtin_amdgcn_global_load_async_to_lds_b128)
#define HAVE_ASYNC_LDS 1
typedef __attribute__((address_space(1))) v4i gv4i;  // global int4
typedef __attribute__((address_space(3))) v4i lv4i;  // LDS int4
#endif

__device__ __forceinline__ v8f wmma4(v2f a, v2f b, v8f c) {
  // D = A(16x4,f32) x B(4x16,f32) + C(16x16,f32)
  return __builtin_amdgcn_wmma_f32_16x16x4_f32(false, a, false, b, (short)0, c,
                                               false, false);
}
__device__ __forceinline__ int imin(int a, int b) { return a < b ? a : b; }

// ---------------- degree / norm ----------------
__global__ void deg_kernel(const int* __restrict__ dst, float* __restrict__ deg,
                           int E) {
  int e = blockIdx.x * blockDim.x + threadIdx.x;
  if (e < E) atomicAdd(&deg[dst[e]], 1.0f);
}

__global__ void inv_sqrt_kernel(float* __restrict__ deg, int n) {
  int i = blockIdx.x * blockDim.x + threadIdx.x;
  if (i < n) deg[i] = rsqrtf(deg[i] + 1.0f);  // +1 = self loop; always > 0
}

// ------------- f32 WMMA GEMM, row-major A[MxK] @ B[KxN], branch-free guards -------------
// ACT: 0 = none, 1 = relu.  NG: true only when N % 16 != 0 (N = 38 case).
// One 16x16 tile per wave, 8 waves per block. Guard-free main K loop + 1 guarded tail.
template <int ACT, bool NG>
__global__ void gemm_wmma_t(const float* __restrict__ A, const float* __restrict__ B,
                            const float* __restrict__ bias, float* __restrict__ C,
                            int M, int N, int K) {
  const int lane = threadIdx.x & 31;
  const int wave = threadIdx.x >> 5;
  const int m0 = (blockIdx.y * 8 + wave) * 16;
  const int n0 = blockIdx.x * 16;
  if (m0 >= M) return;                 // wave-uniform: EXEC all-1 for WMMA
  const int half = lane >> 4;          // lanes 16-31 hold K+2,K+3
  const int l15 = lane & 15;
  const float* __restrict__ Ar = A + (size_t)(m0 + l15) * K;
  const int bcol = n0 + l15;
  const bool bv = !NG || (bcol < N);
  const int bcc = NG ? imin(bcol, N - 1) : bcol;  // clamped, always in-bounds
  const int Kmain = K & ~3;
  v8f acc = {};
#pragma unroll 2
  for (int k0 = 0; k0 < Kmain; k0 += 4) {
    const int ka = k0 + half * 2;
    v2f a, b;
    a.x = Ar[ka];
    a.y = Ar[ka + 1];
    float b0 = B[ka * N + bcc];
    float b1 = B[(ka + 1) * N + bcc];
    b.x = bv ? b0 : 0.0f;
    b.y = bv ? b1 : 0.0f;
    acc = wmma4(a, b, acc);
  }
  if (K & 3) {  // single guarded tail (K = 38): clamp addresses, select values
    const int ka = Kmain + half * 2;
    const int k0c = imin(ka, K - 1), k1c = imin(ka + 1, K - 1);
    const bool v0 = ka < K, v1 = (ka + 1) < K;
    v2f a, b;
    float a0 = Ar[k0c], a1 = Ar[k1c];
    float b0 = B[k0c * N + bcc], b1 = B[k1c * N + bcc];
    a.x = v0 ? a0 : 0.0f;
    a.y = v1 ? a1 : 0.0f;
    b.x = (v0 && bv) ? b0 : 0.0f;
    b.y = (v1 && bv) ? b1 : 0.0f;
    acc = wmma4(a, b, acc);
  }
  if (bv) {
    const float bb = bias ? bias[bcol] : 0.0f;
    float* __restrict__ Cp = C + (size_t)(m0 + half * 8) * N + bcol;
#pragma unroll
    for (int v = 0; v < 8; ++v) {
      float x = acc[v] + bb;
      if (ACT == 1) x = fmaxf(x, 0.0f);
      Cp[(size_t)v * N] = x;
    }
  }
}

// ------- edge scatter (float4): agg[dst] += inv[s]*inv[d] * T[src] -------
__global__ void scatter_edges4(const float4* __restrict__ T, const int* __restrict__ src,
                               const int* __restrict__ dst, const float* __restrict__ inv,
                               float* __restrict__ agg, int E, int F4) {
  long t = (long)blockIdx.x * blockDim.x + threadIdx.x;
  if (t >= (long)E * F4) return;
  int e = (int)(t / F4);
  int f = (int)(t % F4);
  int s = src[e], d = dst[e];
  float w = inv[s] * inv[d];
  float4 v = T[(long)s * F4 + f];
  float* ap = agg + ((long)d * F4 + f) * 4;
  atomicAdd(ap + 0, w * v.x);
  atomicAdd(ap + 1, w * v.y);
  atomicAdd(ap + 2, w * v.z);
  atomicAdd(ap + 3, w * v.w);
}

// ------- out = relu(agg + inv^2 * T + bias), float4, self-loop folded in -------
__global__ void bias_act_selfloop4(const float4* __restrict__ agg,
                                   const float4* __restrict__ T,
                                   const float* __restrict__ inv,
                                   const float4* __restrict__ bias,
                                   float4* __restrict__ out, int n, int F4) {
  long t = (long)blockIdx.x * blockDim.x + threadIdx.x;
  if (t >= (long)n * F4) return;
  int i = (int)(t / F4);
  int f = (int)(t % F4);
  float w = inv[i];
  float w2 = w * w;
  float4 a = agg[t], tt = T[t], b = bias[f];
  float4 r;
  r.x = fmaxf(a.x + w2 * tt.x + b.x, 0.0f);
  r.y = fmaxf(a.y + w2 * tt.y + b.y, 0.0f);
  r.z = fmaxf(a.z + w2 * tt.z + b.z, 0.0f);
  r.w = fmaxf(a.w + w2 * tt.w + b.w, 0.0f);
  out[t] = r;
}

// ---------------- adjacency: sigmoid(EL @ EL^T), K = 64 fixed ----------------
// Block = 8 waves = 128 M-rows x 64 N-cols. B rows staged in LDS (async if available),
// A fragments preloaded once per wave, 4 interleaved accumulator chains,
// non-temporal stores (1 GB output >> 192 MB L2, never re-read).
__global__ void adj_wmma(const float* __restrict__ EL, float* __restrict__ adj, int n) {
  __shared__ float Bs[64 * 64];  // 16 KB
  const int t = threadIdx.x;
  const int lane = t & 31;
  const int wave = t >> 5;
  const int half = lane >> 4;
  const int l15 = lane & 15;
  const int n0base = blockIdx.x * 64;
  const int m0 = (blockIdx.y * 8 + wave) * 16;

  // ---- stage 64 B-rows (EL rows n0base..n0base+63, 64 f32 each) into LDS ----
  const float4* __restrict__ gsrc = (const float4*)(EL + (size_t)n0base * 64);
  float4* ldst = (float4*)Bs;
#ifdef HAVE_ASYNC_LDS
#pragma unroll
  for (int i = 0; i < 4; ++i) {
    __builtin_amdgcn_global_load_async_to_lds_b128(
        (gv4i*)(gsrc + t + i * 256), (lv4i*)(ldst + t + i * 256), 0, 0);
  }
  asm volatile("s_wait_asynccnt 0x0" ::: "memory");
#else
#pragma unroll
  for (int i = 0; i < 4; ++i) ldst[t + i * 256] = gsrc[t + i * 256];
#endif
  __syncthreads();

  // ---- preload the 16 A fragments for this wave's 16 rows ----
  const float2* __restrict__ Ar2 = (const float2*)(EL + (size_t)(m0 + l15) * 64);
  v2f afr[16];
#pragma unroll
  for (int k0 = 0; k0 < 16; ++k0) {
    float2 a = Ar2[k0 * 2 + half];
    afr[k0].x = a.x;
    afr[k0].y = a.y;
  }

  // ---- 4 independent 16x16 accumulator chains over K = 64 ----
  v8f acc[4] = {};
#pragma unroll
  for (int k0 = 0; k0 < 16; ++k0) {
#pragma unroll
    for (int nt = 0; nt < 4; ++nt) {
      const float2* br2 = (const float2*)(Bs + (nt * 16 + l15) * 64);
      float2 bb = br2[k0 * 2 + half];
      v2f b;
      b.x = bb.x;
      b.y = bb.y;
      acc[nt] = wmma4(afr[k0], b, acc[nt]);
    }
  }

  // ---- sigmoid epilogue, non-temporal streaming stores ----
#pragma unroll
  for (int nt = 0; nt < 4; ++nt) {
    const long col = n0base + nt * 16 + l15;
    float* __restrict__ Cp = adj + (long)(m0 + half * 8) * n + col;
#pragma unroll
    for (int v = 0; v < 8; ++v) {
      float x = acc[nt][v];
      float s = 1.0f / (1.0f + __expf(-x));
      __builtin_nontemporal_store(s, Cp + (long)v * n);
    }
  }
}

// ---------------- per-graph segment max via order-preserving uint ----------------
__device__ __forceinline__ unsigned enc_f(float f) {
  unsigned u = __float_as_uint(f);
  return (u & 0x80000000u) ? ~u : (u | 0x80000000u);
}
__device__ __forceinline__ float dec_f(unsigned u) {
  return (u & 0x80000000u) ? __uint_as_float(u & 0x7FFFFFFFu)
                           : __uint_as_float(~u);
}

__global__ void zg_init(unsigned* __restrict__ e, int n) {
  int i = blockIdx.x * blockDim.x + threadIdx.x;
  if (i < n) e[i] = 0x007FFFFFu;  // enc(-inf)
}
__global__ void zg_atomic(const float* __restrict__ z, const int* __restrict__ batch,
                          unsigned* __restrict__ e, int n) {
  long t = (long)blockIdx.x * blockDim.x + threadIdx.x;
  if (t >= (long)n * 64) return;
  int i = (int)(t >> 6);
  int f = (int)(t & 63);
  atomicMax(&e[batch[i] * 64 + f], enc_f(z[t]));
}
__global__ void zg_decode(const unsigned* __restrict__ e, float* __restrict__ out, int n) {
  int i = blockIdx.x * blockDim.x + threadIdx.x;
  if (i < n) out[i] = dec_f(e[i]);
}

// ---------------- driver ----------------
static void launch_gemm(const float* A, const float* B, const float* bias, float* C,
                        int M, int N, int K, int act, hipStream_t s) {
  dim3 g((N + 15) / 16, (M + 127) / 128), b(256);
  const bool ng = (N & 15) != 0;
  if (act) {
    if (ng) gemm_wmma_t<1, true><<<g, b, 0, s>>>(A, B, bias, C, M, N, K);
    else    gemm_wmma_t<1, false><<<g, b, 0, s>>>(A, B, bias, C, M, N, K);
  } else {
    if (ng) gemm_wmma_t<0, true><<<g, b, 0, s>>>(A, B, bias, C, M, N, K);
    else    gemm_wmma_t<0, false><<<g, b, 0, s>>>(A, B, bias, C, M, N, K);
  }
}

extern "C" void kernel_launch(void* const* d_in, const int* in_sizes, int n_in,
                              void* d_out, int out_size, void* d_ws, size_t ws_size,
                              hipStream_t stream) {
  const float* x   = (const float*)d_in[0];
  const int*   ei  = (const int*)d_in[1];
  const int* batch = (const int*)d_in[2];
  const float* Wg0 = (const float*)d_in[3];  const float* bg0 = (const float*)d_in[4];
  const float* Wg1 = (const float*)d_in[5];  const float* bg1 = (const float*)d_in[6];
  const float* Wg2 = (const float*)d_in[7];  const float* bg2 = (const float*)d_in[8];
  const float* Wd0 = (const float*)d_in[9];  const float* bd0 = (const float*)d_in[10];
  const float* Wd1 = (const float*)d_in[11]; const float* bd1 = (const float*)d_in[12];
  const float* Wd2 = (const float*)d_in[13]; const float* bd2 = (const float*)d_in[14];
  const float* We  = (const float*)d_in[15]; const float* be  = (const float*)d_in[16];

  const int* src = ei;       // edge_index[0, :]
  const int* dst = ei + EE;  // edge_index[1, :]

  float* out  = (float*)d_out;
  float* z    = out;                       // 16384 x 64
  float* zg   = z + (size_t)NN * 64;       // 128 x 64
  float* xrec = zg + (size_t)GG * 64;      // 16384 x 38
  float* adj  = xrec + (size_t)NN * NF;    // 16384 x 16384 (written last)

  // Big intermediates live inside adj region (dead before adj kernel runs).
  float* bufA = adj;                       // up to 16384 x 256
  float* bufB = adj + (size_t)NN * 256;    // up to 16384 x 256

  float*    inv = (float*)d_ws;                    // 16384
  unsigned* zge = (unsigned*)(inv + NN);           // 8192
  float*    EL  = (float*)(zge + (size_t)GG * 64); // 16384 x 64

  const dim3 blk(256);

  // ---- GCN normalization ----
  (void)hipMemsetAsync(inv, 0, (size_t)NN * 4, stream);
  deg_kernel<<<(EE + 255) / 256, blk, 0, stream>>>(dst, inv, EE);
  inv_sqrt_kernel<<<(NN + 255) / 256, blk, 0, stream>>>(inv, NN);

  // ---- encoder layer 0: 38 -> 256 ----
  launch_gemm(x, Wg0, nullptr, bufA, NN, 256, NF, 0, stream);
  (void)hipMemsetAsync(bufB, 0, (size_t)NN * 256 * 4, stream);
  scatter_edges4<<<(unsigned)(((long)EE * 64 + 255) / 256), blk, 0, stream>>>(
      (const float4*)bufA, src, dst, inv, bufB, EE, 64);
  bias_act_selfloop4<<<(unsigned)(((long)NN * 64 + 255) / 256), blk, 0, stream>>>(
      (const float4*)bufB, (const float4*)bufA, inv, (const float4*)bg0,
      (float4*)bufA, NN, 64);

  // ---- encoder layer 1: 256 -> 128 ----
  launch_gemm(bufA, Wg1, nullptr, bufB, NN, 128, 256, 0, stream);
  (void)hipMemsetAsync(bufA, 0, (size_t)NN * 128 * 4, stream);
  scatter_edges4<<<(unsigned)(((long)EE * 32 + 255) / 256), blk, 0, stream>>>(
      (const float4*)bufB, src, dst, inv, bufA, EE, 32);
  bias_act_selfloop4<<<(unsigned)(((long)NN * 32 + 255) / 256), blk, 0, stream>>>(
      (const float4*)bufA, (const float4*)bufB, inv, (const float4*)bg1,
      (float4*)bufA, NN, 32);

  // ---- encoder layer 2: 128 -> 64 (z written to d_out) ----
  launch_gemm(bufA, Wg2, nullptr, bufB, NN, 64, 128, 0, stream);
  (void)hipMemsetAsync(bufA, 0, (size_t)NN * 64 * 4, stream);
  scatter_edges4<<<(unsigned)(((long)EE * 16 + 255) / 256), blk, 0, stream>>>(
      (const float4*)bufB, src, dst, inv, bufA, EE, 16);
  bias_act_selfloop4<<<(unsigned)(((long)NN * 16 + 255) / 256), blk, 0, stream>>>(
      (const float4*)bufA, (const float4*)bufB, inv, (const float4*)bg2,
      (float4*)z, NN, 16);

  // ---- decoder MLP: 64 -> 128 -> 256 -> 38 ----
  launch_gemm(z,    Wd0, bd0, bufA, NN, 128, 64, 1, stream);
  launch_gemm(bufA, Wd1, bd1, bufB, NN, 256, 128, 1, stream);
  launch_gemm(bufB, Wd2, bd2, xrec, NN, NF, 256, 0, stream);

  // ---- edge predictor logits: 64 -> 64 ----
  launch_gemm(z, We, be, EL, NN, 64, 64, 0, stream);

  // ---- per-graph max pool ----
  zg_init<<<(GG * 64 + 255) / 256, blk, 0, stream>>>(zge, GG * 64);
  zg_atomic<<<(unsigned)(((long)NN * 64 + 255) / 256), blk, 0, stream>>>(z, batch, zge, NN);
  zg_decode<<<(GG * 64 + 255) / 256, blk, 0, stream>>>(zge, zg, GG * 64);

  // ---- dense adjacency reconstruction (overwrites scratch region fully) ----
  adj_wmma<<<dim3(NN / 64, NN / 128), blk, 0, stream>>>(EL, adj, NN);
}